// CrossInteraction_11682311045657
// MI455X (gfx1250) — compile-verified
//
#include <hip/hip_runtime.h>

typedef __attribute__((ext_vector_type(2))) float v2f;
typedef __attribute__((ext_vector_type(4))) float v4f;
typedef __attribute__((ext_vector_type(8))) float v8f;

#define BATCH 256
#define DIM1  512
#define DIM2  1024
#define OUTW  (DIM1 + DIM2)

// ---------------------------------------------------------------------------
// Kernel 1: per-batch-row means of x1 (512 wide) and x2 (1024 wide).
// One wave (32 threads) per 16 batch rows. Each lane VALU-accumulates a
// contiguous strip of its row into one slot of a 16x4 f32 A-matrix, then a
// single V_WMMA_F32_16X16X4_F32 against an all-ones B-matrix performs the
// cross-K and cross-lane reduction: D[m,n] = rowsum(m) for every n.
// A-matrix f32 16x4 layout (ISA 7.12.2): lanes 0-15 hold K=0 (vgpr0), K=1
// (vgpr1) for rows M=lane; lanes 16-31 hold K=2, K=3 for rows M=lane-16.
// C/D layout: lanes 0-15 -> M=0..7 in c[0..7] (N=lane); lanes 16-31 -> M=8..15.
// ---------------------------------------------------------------------------
__global__ __launch_bounds__(32)
void row_means_wmma(const float* __restrict__ x1, const float* __restrict__ x2,
                    float* __restrict__ m1, float* __restrict__ m2) {
  const int lane  = threadIdx.x;      // 0..31, EXEC all ones (block == 1 wave)
  const int rbase = blockIdx.x * 16;  // 16 rows per wave
  const int row16 = lane & 15;
  const int hi    = lane >> 4;        // 0: K-slots {0,1}, 1: K-slots {2,3}
  const int li    = lane & 15;

  v2f ones; ones.x = 1.0f; ones.y = 1.0f;

  // ---- x1: 512 cols = 4 K-slots x 128 cols ----
  {
    const float* rp = x1 + (size_t)(rbase + row16) * DIM1;
    const float* p0 = rp + (2 * hi)     * 128;
    const float* p1 = rp + (2 * hi + 1) * 128;
    float ax = 0.0f, ay = 0.0f;
#pragma unroll
    for (int t = 0; t < 128; t += 4) {
      const v4f q0 = *(const v4f*)(p0 + t);
      const v4f q1 = *(const v4f*)(p1 + t);
      ax += q0.x + q0.y + q0.z + q0.w;
      ay += q1.x + q1.y + q1.z + q1.w;
    }
    v2f a; a.x = ax; a.y = ay;
    v8f c = {};
    // D = A x ones + 0  ->  D[m, n] = rowsum(m), replicated over n
    c = __builtin_amdgcn_wmma_f32_16x16x4_f32(false, a, false, ones,
                                              (short)0, c, false, false);
    if (li == 0) {  // lane 0 -> rows 0..7, lane 16 -> rows 8..15
#pragma unroll
      for (int r = 0; r < 8; ++r)
        m1[rbase + hi * 8 + r] = c[r] * (1.0f / DIM1);
    }
  }

  // ---- x2: 1024 cols = 4 K-slots x 256 cols ----
  {
    const float* rp = x2 + (size_t)(rbase + row16) * DIM2;
    const float* p0 = rp + (2 * hi)     * 256;
    const float* p1 = rp + (2 * hi + 1) * 256;
    float ax = 0.0f, ay = 0.0f;
#pragma unroll
    for (int t = 0; t < 256; t += 4) {
      const v4f q0 = *(const v4f*)(p0 + t);
      const v4f q1 = *(const v4f*)(p1 + t);
      ax += q0.x + q0.y + q0.z + q0.w;
      ay += q1.x + q1.y + q1.z + q1.w;
    }
    v2f a; a.x = ax; a.y = ay;
    v8f c = {};
    c = __builtin_amdgcn_wmma_f32_16x16x4_f32(false, a, false, ones,
                                              (short)0, c, false, false);
    if (li == 0) {
#pragma unroll
      for (int r = 0; r < 8; ++r)
        m2[rbase + hi * 8 + r] = c[r] * (1.0f / DIM2);
    }
  }
}

// ---------------------------------------------------------------------------
// Kernel 2: out[b, 0:512]      = x1[b,:] * mean(x2[b,:])
//           out[b, 512:1536]   = x2[b,:] * mean(x1[b,:])
// Fully coalesced reads and writes; mean vectors broadcast from L2.
// ---------------------------------------------------------------------------
__global__ __launch_bounds__(256)
void scale_concat(const float* __restrict__ x1, const float* __restrict__ x2,
                  const float* __restrict__ m1, const float* __restrict__ m2,
                  float* __restrict__ out) {
  const int idx = blockIdx.x * blockDim.x + threadIdx.x;
  if (idx >= BATCH * OUTW) return;
  const int b = idx / OUTW;
  const int c = idx - b * OUTW;
  float v;
  if (c < DIM1) v = x1[b * DIM1 + c]          * m2[b];
  else          v = x2[b * DIM2 + (c - DIM1)] * m1[b];
  out[idx] = v;
}

extern "C" void kernel_launch(void* const* d_in, const int* in_sizes, int n_in,
                              void* d_out, int out_size, void* d_ws, size_t ws_size,
                              hipStream_t stream) {
  const float* x1 = (const float*)d_in[0];   // (256, 512)  f32
  const float* x2 = (const float*)d_in[1];   // (256, 1024) f32
  float* out = (float*)d_out;                // (256, 1536) f32

  float* m1 = (float*)d_ws;                  // [BATCH] row-means of x1
  float* m2 = m1 + BATCH;                    // [BATCH] row-means of x2

  row_means_wmma<<<BATCH / 16, 32, 0, stream>>>(x1, x2, m1, m2);

  const int total  = BATCH * OUTW;           // 393216
  const int tpb    = 256;
  const int blocks = (total + tpb - 1) / tpb;
  scale_concat<<<blocks, tpb, 0, stream>>>(x1, x2, m1, m2, out);
}